// NaiveE2V_10290741641949
// MI455X (gfx1250) — compile-verified
//
#include <hip/hip_runtime.h>
#include <hip/hip_bf16.h>

typedef __attribute__((ext_vector_type(2))) float v2f;
typedef __attribute__((ext_vector_type(8))) float v8f;

#define NV 4000
#define EE 16000
#define DD 64
#define KSPLIT 8
#define KCHUNK (EE / KSPLIT)   // 2000 -> 500 WMMA k-steps -> 125 unrolled groups
#define NTILES (NV / 16)       // 250

// ---------------------------------------------------------------------------
// Kernel 1: x1eT[j, e] = (x_e[e,:] @ W[1, order[e]])[j]   (TRANSPOSED B operand)
// Transposed layout makes each WMMA B-fragment a contiguous 8-byte pair.
// ---------------------------------------------------------------------------
__global__ __launch_bounds__(64) void edge_x1_kernel(const float* __restrict__ xe,
                                                     const int* __restrict__ ord,
                                                     const float* __restrict__ W,
                                                     float* __restrict__ x1eT) {
    const int e = blockIdx.x;
    const int j = threadIdx.x;
    __shared__ float xs[DD];
    xs[j] = xe[(size_t)e * DD + j];
    __syncthreads();
    const int k = ord[e];
    const float* Wm = W + ((size_t)(5 + k)) * (DD * DD);   // W[1, k]
    float acc = 0.f;
#pragma unroll 8
    for (int i = 0; i < DD; ++i) acc = fmaf(xs[i], Wm[i * DD + j], acc);
    x1eT[(size_t)j * EE + e] = acc;    // transposed store
}

// ---------------------------------------------------------------------------
// Kernel 2: deterministic column sums.
// sums[k][j] = sum over edges with order==k of x_e[e,j]   (k = 0..4)
// sums[5][j] = sum over vertices of x_v[n,j]
// ---------------------------------------------------------------------------
__global__ __launch_bounds__(256) void colsum_kernel(const float* __restrict__ xv,
                                                     const float* __restrict__ xe,
                                                     const int* __restrict__ ord,
                                                     float* __restrict__ sums) {
    const int j = blockIdx.x;       // column 0..63
    const int t = threadIdx.x;
    float a0 = 0, a1 = 0, a2 = 0, a3 = 0, a4 = 0, av = 0;
    for (int e = t; e < EE; e += 256) {
        const float v = xe[(size_t)e * DD + j];
        const int k = ord[e];
        a0 += (k == 0) ? v : 0.f;
        a1 += (k == 1) ? v : 0.f;
        a2 += (k == 2) ? v : 0.f;
        a3 += (k == 3) ? v : 0.f;
        a4 += (k == 4) ? v : 0.f;
    }
    for (int n = t; n < NV; n += 256) av += xv[(size_t)n * DD + j];

    __shared__ float red[256];
    float vals[6] = {a0, a1, a2, a3, a4, av};
#pragma unroll
    for (int s = 0; s < 6; ++s) {
        __syncthreads();
        red[t] = vals[s];
        __syncthreads();
        for (int w = 128; w > 0; w >>= 1) {
            if (t < w) red[t] += red[t + w];
            __syncthreads();
        }
        if (t == 0) sums[s * DD + j] = red[0];
    }
}

// ---------------------------------------------------------------------------
// Kernel 3: x0[j] = ( colsum(x_v) @ W[0,1] + sum_k sums[k] @ W[0,k] ) / (N+E)
// ---------------------------------------------------------------------------
__global__ __launch_bounds__(64) void x0_kernel(const float* __restrict__ W,
                                                const float* __restrict__ sums,
                                                float* __restrict__ x0) {
    const int j = threadIdx.x;
    float acc = 0.f;
    const float* W01 = W + 1 * (DD * DD);   // W[0,1]
#pragma unroll 4
    for (int i = 0; i < DD; ++i) acc = fmaf(sums[5 * DD + i], W01[i * DD + j], acc);
    for (int k = 0; k < 5; ++k) {
        const float* W0k = W + (size_t)k * (DD * DD);   // W[0,k]
#pragma unroll 4
        for (int i = 0; i < DD; ++i) acc = fmaf(sums[k * DD + i], W0k[i * DD + j], acc);
    }
    x0[j] = acc / (float)(NV + EE);
}

// ---------------------------------------------------------------------------
// Kernel 4: WMMA fp32 GEMM with streamed pointers.
// partial[tile, split, 16, 64] = incidence[tile rows, K-chunk] @ x1_e[K-chunk, :]
//                                (+ x_v[tile rows,:] @ W[1,1] when split==0)
// One wave per block: 16(M) x 64(N) tile = 4 accumulators of 8 VGPRs.
//
// V_WMMA_F32_16X16X4_F32 operand layout (ISA 7.12.2):
//  A (16x4):  lanes 0-15 -> rows M=lane, VGPR{0,1}=K{0,1};
//             lanes 16-31 -> rows M=lane-16, VGPR{0,1}=K{2,3}
//  B (4x16):  VGPR0: lanes 0-15 = row K0, lanes 16-31 = row K2;
//             VGPR1: lanes 0-15 = row K1, lanes 16-31 = row K3 (cols = lane&15)
//  C/D:       VGPR v: lanes 0-15 -> M=v, lanes 16-31 -> M=v+8, N = lane&15
// With x1eT[col][e], fragment {B[k+2h][col], B[k+2h+1][col]} is one b64 load.
// ---------------------------------------------------------------------------
__global__ __launch_bounds__(32) void gemm_kernel(const float* __restrict__ inc,
                                                  const float* __restrict__ x1eT,
                                                  const float* __restrict__ xv,
                                                  const float* __restrict__ W,
                                                  float* __restrict__ partial) {
    const int tile  = blockIdx.x;   // 0..249
    const int split = blockIdx.y;   // 0..7
    const int lane  = threadIdx.x;  // 0..31
    const int half  = lane >> 4;    // 0 or 1
    const int m     = lane & 15;
    const int row0  = tile * 16;

    v8f acc[4];
#pragma unroll
    for (int t = 0; t < 4; ++t) acc[t] = (v8f){0.f, 0.f, 0.f, 0.f, 0.f, 0.f, 0.f, 0.f};

    const int k0 = split * KCHUNK;
    const int lo = 2 * half;   // 0 or 2
    const float* ap  = inc  + (size_t)(row0 + m) * EE + lo + k0;
    const float* bp0 = x1eT + (size_t)( 0 + m) * EE + lo + k0;
    const float* bp1 = x1eT + (size_t)(16 + m) * EE + lo + k0;
    const float* bp2 = x1eT + (size_t)(32 + m) * EE + lo + k0;
    const float* bp3 = x1eT + (size_t)(48 + m) * EE + lo + k0;

    for (int it = 0; it < KCHUNK / 16; ++it) {   // 4 WMMA k-steps per group
        __builtin_prefetch(ap + 512, 0, 1);      // 2KB ahead on the cold incidence stream
#pragma unroll
        for (int u = 0; u < 4; ++u) {            // immediate offsets 0/16/32/48 B
            const v2f a  = *(const v2f*)(ap  + 4 * u);
            const v2f b0 = *(const v2f*)(bp0 + 4 * u);
            const v2f b1 = *(const v2f*)(bp1 + 4 * u);
            const v2f b2 = *(const v2f*)(bp2 + 4 * u);
            const v2f b3 = *(const v2f*)(bp3 + 4 * u);
            acc[0] = __builtin_amdgcn_wmma_f32_16x16x4_f32(false, a, false, b0, (short)0, acc[0], false, false);
            acc[1] = __builtin_amdgcn_wmma_f32_16x16x4_f32(false, a, false, b1, (short)0, acc[1], false, false);
            acc[2] = __builtin_amdgcn_wmma_f32_16x16x4_f32(false, a, false, b2, (short)0, acc[2], false, false);
            acc[3] = __builtin_amdgcn_wmma_f32_16x16x4_f32(false, a, false, b3, (short)0, acc[3], false, false);
        }
        ap += 16; bp0 += 16; bp1 += 16; bp2 += 16; bp3 += 16;
    }

    if (split == 0) {
        // fold in x_v[rows,:] @ W[1,1]   (K = 64 -> 16 more WMMA steps)
        const float* W11 = W + (size_t)(5 + 1) * (DD * DD);   // W[1,1] row-major
        const float* av = xv + (size_t)(row0 + m) * DD + lo;
#pragma unroll 4
        for (int k = 0; k < DD; k += 4) {
            const v2f a = *(const v2f*)(av + k);
            const float* bp = W11 + (size_t)(k + lo) * DD + m;
#pragma unroll
            for (int t = 0; t < 4; ++t) {
                v2f b;
                b.x = bp[t * 16];
                b.y = bp[DD + t * 16];
                acc[t] = __builtin_amdgcn_wmma_f32_16x16x4_f32(
                    false, a, false, b, (short)0, acc[t], false, false);
            }
        }
    }

    // store partial tile: [(tile*KSPLIT + split)][16][64]
    float* out = partial + (size_t)(tile * KSPLIT + split) * 16 * DD;
#pragma unroll
    for (int t = 0; t < 4; ++t) {
#pragma unroll
        for (int v = 0; v < 8; ++v) {
            const int r = v + 8 * half;
            out[r * DD + t * 16 + m] = acc[t][v];
        }
    }
}

// ---------------------------------------------------------------------------
// Kernel 5: combine K-split partials + epilogue
// out[n,j] = x0[j] + b[j] + (sum_splits partial)/(1 + suffix[n])
// ---------------------------------------------------------------------------
__global__ __launch_bounds__(256) void final_kernel(const float* __restrict__ partial,
                                                    const float* __restrict__ x0,
                                                    const float* __restrict__ b,
                                                    const float* __restrict__ suffix,
                                                    float* __restrict__ out) {
    const int idx = blockIdx.x * 256 + threadIdx.x;   // < 256000
    const int row = idx >> 6;
    const int col = idx & 63;
    const int tile = row >> 4;
    const int rl = row & 15;
    const float* p = partial + ((size_t)tile * KSPLIT * 16 + rl) * DD + col;
    float s = 0.f;
#pragma unroll
    for (int sp = 0; sp < KSPLIT; ++sp) s += p[(size_t)sp * 16 * DD];
    out[idx] = x0[col] + b[col] + s / (1.0f + suffix[row]);
}

// ---------------------------------------------------------------------------
extern "C" void kernel_launch(void* const* d_in, const int* in_sizes, int n_in,
                              void* d_out, int out_size, void* d_ws, size_t ws_size,
                              hipStream_t stream) {
    const float* x_v    = (const float*)d_in[0];
    const float* x_e    = (const float*)d_in[1];
    const float* inc    = (const float*)d_in[2];
    const int*   ord    = (const int*)d_in[3];
    const float* suffix = (const float*)d_in[4];
    const float* W      = (const float*)d_in[5];
    const float* b      = (const float*)d_in[6];
    float* out = (float*)d_out;

    // workspace layout (floats)
    float* ws      = (float*)d_ws;
    float* x1eT    = ws;                               // D*E            = 1,024,000
    float* partial = x1eT + (size_t)DD * EE;           // 250*8*16*64    = 2,048,000
    float* sums    = partial + (size_t)NTILES * KSPLIT * 16 * DD;   // 6*64
    float* x0      = sums + 6 * DD;                    // 64

    edge_x1_kernel<<<EE, 64, 0, stream>>>(x_e, ord, W, x1eT);
    colsum_kernel<<<DD, 256, 0, stream>>>(x_v, x_e, ord, sums);
    x0_kernel<<<1, 64, 0, stream>>>(W, sums, x0);
    gemm_kernel<<<dim3(NTILES, KSPLIT), 32, 0, stream>>>(inc, x1eT, x_v, W, partial);
    final_kernel<<<(NV * DD) / 256, 256, 0, stream>>>(partial, x0, b, suffix, out);
}